// RNN_28166395527530
// MI455X (gfx1250) — compile-verified
//
#include <hip/hip_runtime.h>
#include <cmath>

#define Bdim 16
#define Sdim 512
#define Edim 300
#define Hdim 512
#define Vdim 10000

typedef __attribute__((ext_vector_type(2))) float v2f;
typedef __attribute__((ext_vector_type(8))) float v8f;
typedef __attribute__((ext_vector_type(4))) unsigned int v4u;
typedef __attribute__((ext_vector_type(4))) int v4i;
typedef __attribute__((ext_vector_type(8))) int v8i;

// --- TDM availability (device pass only; host pass uses fallback) ----------
#if defined(__gfx1250__) && defined(__has_builtin)
#if __has_builtin(__builtin_amdgcn_tensor_load_to_lds) && \
    __has_builtin(__builtin_amdgcn_s_wait_tensorcnt)
#define HAVE_TDM 1
#endif
#endif
#ifndef HAVE_TDM
#define HAVE_TDM 0
#endif

__device__ __forceinline__ v8f wmma_f32_16x16x4(v2f a, v2f b, v8f c) {
  return __builtin_amdgcn_wmma_f32_16x16x4_f32(false, a, false, b, (short)0, c,
                                               false, false);
}

#if HAVE_TDM
__device__ __forceinline__ unsigned int ldsaddr(const void* p) {
  // flat LDS aperture: low 32 bits = LDS byte offset
  return (unsigned int)(unsigned long long)(uintptr_t)p;
}

// 2D tile DMA: global (row-major, stride0 elems of 4B) -> LDS, with hardware
// row padding of 1 DWORD every (1<<(pad_code+1)) DWORDs (odd LDS stride ->
// bank-conflict-free column reads). OOB rows (y >= tensor_d1) read as zero.
__device__ __forceinline__ void tdm_load_2d(unsigned int lds_off,
                                            const void* gptr,
                                            unsigned int tile_d0,
                                            unsigned int tile_d1,
                                            unsigned int tensor_d0,
                                            unsigned int tensor_d1,
                                            unsigned long long stride0,
                                            unsigned int pad_code) {
  unsigned long long ga = (unsigned long long)(uintptr_t)gptr;
  v4u g0;
  g0.x = 1u;                                    // count=1: valid user D#
  g0.y = lds_off;                               // LDS byte address
  g0.z = (unsigned int)ga;                      // global_addr[31:0]
  g0.w = (unsigned int)((ga >> 32) & 0x01ffffffull) | (2u << 30);  // type=2
  v8i g1;
  g1[0] = (int)((2u << 16)            // data_size = 4B
                | (1u << 20)          // pad_enable
                | (pad_code << 22));  // pad_interval; pad_amount=0 (1 DWORD)
  g1[1] = (int)((tensor_d0 & 0xffffu) << 16);   // atomic_barrier=0 | td0.lo16
  g1[2] = (int)((tensor_d0 >> 16) | ((tensor_d1 & 0xffffu) << 16));
  g1[3] = (int)((tensor_d1 >> 16) | (tile_d0 << 16));
  g1[4] = (int)(tile_d1 & 0xffffu);             // tile_dim1 | tile_dim2=0
  g1[5] = (int)(unsigned int)stride0;           // tensor_dim0_stride[31:0]
  g1[6] = (int)(unsigned int)((stride0 >> 32) & 0xffffull);
  g1[7] = 0;
  v4i gz = {};
#if __clang_major__ >= 23
  v8i gz8 = {};
  __builtin_amdgcn_tensor_load_to_lds(g0, g1, gz, gz, gz8, 0);
#else
  __builtin_amdgcn_tensor_load_to_lds(g0, g1, gz, gz, 0);
#endif
}
#endif  // HAVE_TDM

// ---------------------------------------------------------------------------
// Kernel 1: xw[m,n] = sum_k x[m,k]*W_ih[n,k] + b_ih[n] + b_hh[n]
//   M=8192, K=300, N=512. 64x64 tile, 4 waves. (Tiny: 2.5 GFLOP, manual LDS.)
// ---------------------------------------------------------------------------
__global__ __launch_bounds__(128) void proj_kernel(
    const float* __restrict__ x, const float* __restrict__ W_ih,
    const float* __restrict__ b_ih, const float* __restrict__ b_hh,
    float* __restrict__ xw) {
  __shared__ float As[64][101];
  __shared__ float Bs[64][101];
  const int n0 = blockIdx.x * 64;
  const int m0 = blockIdx.y * 64;
  const int tid = threadIdx.x;
  const int w = tid >> 5, lane = tid & 31, lh = lane & 15, hi = (lane >> 4) & 1;

  v8f acc[4] = {};
  for (int k0 = 0; k0 < Edim; k0 += 100) {
    __syncthreads();
    for (int idx = tid; idx < 64 * 100; idx += 128) {
      int r = idx / 100, c = idx % 100;
      As[r][c] = x[(size_t)(m0 + r) * Edim + k0 + c];
      Bs[r][c] = W_ih[(size_t)(n0 + r) * Edim + k0 + c];
    }
    __syncthreads();
    for (int kk = 0; kk < 100; kk += 4) {
      v2f a;
      a.x = As[16 * w + lh][kk + 2 * hi];
      a.y = As[16 * w + lh][kk + 2 * hi + 1];
#pragma unroll
      for (int nt = 0; nt < 4; ++nt) {
        v2f b;
        b.x = Bs[16 * nt + lh][kk + 2 * hi];
        b.y = Bs[16 * nt + lh][kk + 2 * hi + 1];
        acc[nt] = wmma_f32_16x16x4(a, b, acc[nt]);
      }
    }
  }
#pragma unroll
  for (int nt = 0; nt < 4; ++nt) {
    int col = n0 + 16 * nt + lh;
    float bias = b_ih[col] + b_hh[col];
#pragma unroll
    for (int r = 0; r < 8; ++r) {
      int row = m0 + 16 * w + r + 8 * hi;
      xw[(size_t)row * Hdim + col] = acc[nt][r] + bias;
    }
  }
}

// ---------------------------------------------------------------------------
// Kernel 2: h_t = tanh(xw[:,t,:] + h_{t-1} @ W_hh^T)
//   M=16, N=512, K=512. 8 WGs x 4 waves. W_hh tiles DMA'd by TDM with
//   double-buffered LDS (pad 1 DWORD / 64 -> stride 65, conflict-free).
// ---------------------------------------------------------------------------
__global__ __launch_bounds__(128) void step_kernel(
    float* __restrict__ hs, const float* __restrict__ xw,
    const float* __restrict__ W_hh, int t) {
  __shared__ float hA[16][513];
#if HAVE_TDM
  __shared__ float Ws[2][64][65];
#else
  __shared__ float Ws[1][64][65];
#endif
  const int nbase = blockIdx.x * 64;
  const int tid = threadIdx.x;
  const int w = tid >> 5, lane = tid & 31, lh = lane & 15, hi = (lane >> 4) & 1;

  for (int idx = tid; idx < 16 * Hdim; idx += 128) {
    int bb = idx >> 9, k = idx & 511;
    hA[bb][k] = (t == 0) ? 0.f : hs[((size_t)bb * Sdim + (t - 1)) * Hdim + k];
  }

  v8f acc = {};
  const int NC = Hdim / 64;
#if HAVE_TDM
  if (tid < 32)
    tdm_load_2d(ldsaddr(&Ws[0][0][0]), W_hh + (size_t)nbase * Hdim,
                /*tile*/ 64, 64, /*tensor*/ 512, 512, Hdim, /*pad@64dw*/ 5u);
  for (int c = 0; c < NC; ++c) {
    const int buf = c & 1;
    if (tid < 32) __builtin_amdgcn_s_wait_tensorcnt((short)0);  // chunk c done
    __syncthreads();
    if (c + 1 < NC && tid < 32)  // prefetch next chunk during compute
      tdm_load_2d(ldsaddr(&Ws[buf ^ 1][0][0]),
                  W_hh + (size_t)nbase * Hdim + (size_t)(c + 1) * 64,
                  64, 64, 512, 512, Hdim, 5u);
    const int k0 = c * 64;
    for (int kk = 0; kk < 64; kk += 4) {
      v2f a, b;
      a.x = hA[lh][k0 + kk + 2 * hi];
      a.y = hA[lh][k0 + kk + 2 * hi + 1];
      b.x = Ws[buf][16 * w + lh][kk + 2 * hi];
      b.y = Ws[buf][16 * w + lh][kk + 2 * hi + 1];
      acc = wmma_f32_16x16x4(a, b, acc);
    }
    __syncthreads();
  }
#else
  for (int c = 0; c < NC; ++c) {
    const int k0 = c * 64;
    __syncthreads();
    for (int idx = tid; idx < 64 * 64; idx += 128) {
      int n = idx >> 6, k = idx & 63;
      Ws[0][n][k] = W_hh[(size_t)(nbase + n) * Hdim + k0 + k];
    }
    __syncthreads();
    for (int kk = 0; kk < 64; kk += 4) {
      v2f a, b;
      a.x = hA[lh][k0 + kk + 2 * hi];
      a.y = hA[lh][k0 + kk + 2 * hi + 1];
      b.x = Ws[0][16 * w + lh][kk + 2 * hi];
      b.y = Ws[0][16 * w + lh][kk + 2 * hi + 1];
      acc = wmma_f32_16x16x4(a, b, acc);
    }
  }
#endif
  const int col = nbase + 16 * w + lh;
#pragma unroll
  for (int r = 0; r < 8; ++r) {
    int brow = r + 8 * hi;
    size_t off = ((size_t)brow * Sdim + t) * Hdim + col;
    hs[off] = tanhf(acc[r] + xw[off]);
  }
}

// ---------------------------------------------------------------------------
// Kernel 3: logits[b,:,vtile] = hs[b] @ W_fc[vtile]^T + b_fc, softmax over S,
//   fused. One WG per (vtile,b): M=S=512, N=64, K=512; 32 waves. A/B tiles
//   DMA'd by TDM (pad 1 DWORD / 32 -> stride 33), double-buffered; tail rows
//   of W_fc zero-filled by TDM OOB semantics.
// ---------------------------------------------------------------------------
__device__ __forceinline__ void fc_inner(const float (*Asb)[33],
                                         const float (*Bsb)[33], v8f acc[4],
                                         int w, int lh, int hi) {
  for (int kk = 0; kk < 32; kk += 4) {
    v2f a;
    a.x = Asb[16 * w + lh][kk + 2 * hi];
    a.y = Asb[16 * w + lh][kk + 2 * hi + 1];
#pragma unroll
    for (int nt = 0; nt < 4; ++nt) {
      v2f bb;
      bb.x = Bsb[16 * nt + lh][kk + 2 * hi];
      bb.y = Bsb[16 * nt + lh][kk + 2 * hi + 1];
      acc[nt] = wmma_f32_16x16x4(a, bb, acc[nt]);
    }
  }
}

__global__ __launch_bounds__(1024) void fc_softmax_kernel(
    const float* __restrict__ hs, const float* __restrict__ W_fc,
    const float* __restrict__ b_fc, float* __restrict__ out) {
#if HAVE_TDM
  __shared__ float As[2][512][33];
  __shared__ float Bs[2][64][33];
#else
  __shared__ float As[1][512][33];
  __shared__ float Bs[1][64][33];
#endif
  __shared__ float red[64][32];
  __shared__ float colmax[64];
  __shared__ float colsum[64];
  const int v0 = blockIdx.x * 64;
  const int b = blockIdx.y;
  const int tid = threadIdx.x;
  const int w = tid >> 5, lane = tid & 31, lh = lane & 15, hi = (lane >> 4) & 1;

  v8f acc[4] = {};
  const int NC = Hdim / 32;
#if HAVE_TDM
  const float* gA0 = hs + ((size_t)b * Sdim) * Hdim;
  const float* gB0 = W_fc + (size_t)v0 * Hdim;
  const unsigned int vrows = (unsigned int)(Vdim - v0);  // OOB rows -> zero
  if (tid < 32) {
    tdm_load_2d(ldsaddr(&As[0][0][0]), gA0, 32, 512, 512, 512, Hdim, 4u);
    tdm_load_2d(ldsaddr(&Bs[0][0][0]), gB0, 32, 64, 512, vrows, Hdim, 4u);
  }
  for (int c = 0; c < NC; ++c) {
    const int buf = c & 1;
    if (tid < 32) __builtin_amdgcn_s_wait_tensorcnt((short)0);
    __syncthreads();
    if (c + 1 < NC && tid < 32) {
      tdm_load_2d(ldsaddr(&As[buf ^ 1][0][0]), gA0 + (size_t)(c + 1) * 32, 32,
                  512, 512, 512, Hdim, 4u);
      tdm_load_2d(ldsaddr(&Bs[buf ^ 1][0][0]), gB0 + (size_t)(c + 1) * 32, 32,
                  64, 512, vrows, Hdim, 4u);
    }
    fc_inner(As[buf], Bs[buf], acc, w, lh, hi);
    __syncthreads();
  }
#else
  for (int c = 0; c < NC; ++c) {
    const int k0 = c * 32;
    __syncthreads();
    for (int idx = tid; idx < 512 * 32; idx += 1024) {
      int s = idx >> 5, k = idx & 31;
      As[0][s][k] = hs[((size_t)b * Sdim + s) * Hdim + k0 + k];
    }
    for (int idx = tid; idx < 64 * 32; idx += 1024) {
      int n = idx >> 5, k = idx & 31;
      int vr = v0 + n;
      if (vr >= Vdim) vr = Vdim - 1;
      Bs[0][n][k] = W_fc[(size_t)vr * Hdim + k0 + k];
    }
    __syncthreads();
    fc_inner(As[0], Bs[0], acc, w, lh, hi);
  }
#endif

  // + b_fc (per column)
#pragma unroll
  for (int nt = 0; nt < 4; ++nt) {
    int vc = v0 + 16 * nt + lh;
    if (vc >= Vdim) vc = Vdim - 1;
    float bf = b_fc[vc];
#pragma unroll
    for (int r = 0; r < 8; ++r) acc[nt][r] += bf;
  }

  // ---- softmax over S (rows), per column ----
#pragma unroll
  for (int nt = 0; nt < 4; ++nt) {
    float m = acc[nt][0];
#pragma unroll
    for (int r = 1; r < 8; ++r) m = fmaxf(m, acc[nt][r]);
    m = fmaxf(m, __shfl_xor(m, 16, 32));  // wave32: lanes l <-> l^16
    if (lane < 16) red[16 * nt + lh][w] = m;
  }
  __syncthreads();
  if (tid < 64) {
    float m = red[tid][0];
    for (int i = 1; i < 32; ++i) m = fmaxf(m, red[tid][i]);
    colmax[tid] = m;
  }
  __syncthreads();
#pragma unroll
  for (int nt = 0; nt < 4; ++nt) {
    float cm = colmax[16 * nt + lh];
    float s = 0.f;
#pragma unroll
    for (int r = 0; r < 8; ++r) s += expf(acc[nt][r] - cm);
    s += __shfl_xor(s, 16, 32);
    if (lane < 16) red[16 * nt + lh][w] = s;
  }
  __syncthreads();
  if (tid < 64) {
    float s = 0.f;
    for (int i = 0; i < 32; ++i) s += red[tid][i];
    colsum[tid] = s;
  }
  __syncthreads();
#pragma unroll
  for (int nt = 0; nt < 4; ++nt) {
    int vc = v0 + 16 * nt + lh;
    if (vc < Vdim) {
      float cm = colmax[16 * nt + lh];
      float inv = 1.f / colsum[16 * nt + lh];
#pragma unroll
      for (int r = 0; r < 8; ++r) {
        int srow = 16 * w + r + 8 * hi;
        out[((size_t)b * Sdim + srow) * Vdim + vc] = expf(acc[nt][r] - cm) * inv;
      }
    }
  }
}

// ---------------------------------------------------------------------------
extern "C" void kernel_launch(void* const* d_in, const int* in_sizes, int n_in,
                              void* d_out, int out_size, void* d_ws,
                              size_t ws_size, hipStream_t stream) {
  const float* x    = (const float*)d_in[0];
  const float* W_ih = (const float*)d_in[1];
  const float* W_hh = (const float*)d_in[2];
  const float* b_ih = (const float*)d_in[3];
  const float* b_hh = (const float*)d_in[4];
  const float* W_fc = (const float*)d_in[5];
  const float* b_fc = (const float*)d_in[6];

  float* xw = (float*)d_ws;                     // [B*S, H]  (16 MB)
  float* hs = xw + (size_t)Bdim * Sdim * Hdim;  // [B, S, H] (16 MB)

  proj_kernel<<<dim3(Hdim / 64, (Bdim * Sdim) / 64), 128, 0, stream>>>(
      x, W_ih, b_ih, b_hh, xw);

  for (int t = 0; t < Sdim; ++t)
    step_kernel<<<dim3(Hdim / 64), 128, 0, stream>>>(hs, xw, W_hh, t);

  fc_softmax_kernel<<<dim3((Vdim + 63) / 64, Bdim), 1024, 0, stream>>>(
      hs, W_fc, b_fc, (float*)d_out);
}